// TFDistributedEmbedding_76828374991710
// MI455X (gfx1250) — compile-verified
//
#include <hip/hip_runtime.h>
#include <hip/hip_bf16.h>
#include <stdint.h>

// out[i, :] = emb_weights[inputs_flat[i], :]   (reference's sort/unsort is an identity)
// Pure HBM-bandwidth gather: ~0.5 GiB moved -> ~22 us floor at 23.3 TB/s.

typedef float __attribute__((ext_vector_type(4))) f32x4;
typedef int   __attribute__((ext_vector_type(4))) i32x4;

#define EMBV 32u           // 128 floats per row = 32 float4 = 512 B = one wave32 of b128
#define WAVES_PER_BLOCK 8  // 256 threads
#define ROWS_PER_WAVE 8    // 8 independent b128 gathers in flight per wave
#define ROWS_PER_BLOCK (WAVES_PER_BLOCK * ROWS_PER_WAVE)   // 64

// ---------------------------------------------------------------------------
// Fast path: requires nRows % ROWS_PER_BLOCK == 0 (true for 524288).
// Branch-free; all indexing in unsigned 32-bit so the table (512 MB) and
// output (256 MB) are addressed as SGPR-base + 32-bit VGPR offset, and the 8
// consecutive output stores share one address VGPR via imm offset:512*k.
// ---------------------------------------------------------------------------
__global__ __launch_bounds__(256) void emb_gather_fast(
    const int* __restrict__ ids,
    const f32x4* __restrict__ tab,
    f32x4* __restrict__ out)
{
    const unsigned lane = threadIdx.x & 31u;    // float4 column within the row
    const unsigned wave = threadIdx.x >> 5;
    const unsigned row0 = ((unsigned)blockIdx.x * WAVES_PER_BLOCK + wave) * ROWS_PER_WAVE;

    // Prefetch the sequential id stream ahead (lowers to global_prefetch_b8).
    __builtin_prefetch(ids + row0 + 32 * ROWS_PER_BLOCK, 0, 3);

    // Wave-uniform id fetch: two b128 loads grab all 8 ids (16B aligned).
    i32x4 a = *(const i32x4*)(ids + row0);
    i32x4 b = *(const i32x4*)(ids + row0 + 4);

    // Issue all 8 random-row b128 gathers before any store (latency overlap).
    // 32-bit unsigned indices -> zext once -> saddr + voffset addressing.
    f32x4 x0 = tab[(unsigned)a.x * EMBV + lane];
    f32x4 x1 = tab[(unsigned)a.y * EMBV + lane];
    f32x4 x2 = tab[(unsigned)a.z * EMBV + lane];
    f32x4 x3 = tab[(unsigned)a.w * EMBV + lane];
    f32x4 x4 = tab[(unsigned)b.x * EMBV + lane];
    f32x4 x5 = tab[(unsigned)b.y * EMBV + lane];
    f32x4 x6 = tab[(unsigned)b.z * EMBV + lane];
    f32x4 x7 = tab[(unsigned)b.w * EMBV + lane];

    // Streaming output: non-temporal b128 stores keep L2 free for table rows.
    // One shared 32-bit offset; consecutive rows reached via imm offsets.
    const unsigned o = row0 * EMBV + lane;
    __builtin_nontemporal_store(x0, &out[o + 0u * EMBV]);
    __builtin_nontemporal_store(x1, &out[o + 1u * EMBV]);
    __builtin_nontemporal_store(x2, &out[o + 2u * EMBV]);
    __builtin_nontemporal_store(x3, &out[o + 3u * EMBV]);
    __builtin_nontemporal_store(x4, &out[o + 4u * EMBV]);
    __builtin_nontemporal_store(x5, &out[o + 5u * EMBV]);
    __builtin_nontemporal_store(x6, &out[o + 6u * EMBV]);
    __builtin_nontemporal_store(x7, &out[o + 7u * EMBV]);
}

// ---------------------------------------------------------------------------
// Generic fallback: one row per wave, fully bounds-checked. Only launched when
// nRows is not a multiple of ROWS_PER_BLOCK (never for the reference shapes).
// ---------------------------------------------------------------------------
__global__ __launch_bounds__(256) void emb_gather_generic(
    const int* __restrict__ ids,
    const f32x4* __restrict__ tab,
    f32x4* __restrict__ out,
    int nRows)
{
    const unsigned lane = threadIdx.x & 31u;
    const unsigned wave = threadIdx.x >> 5;
    const int row = (int)(blockIdx.x * WAVES_PER_BLOCK + wave);
    if (row >= nRows) return;
    const unsigned id = (unsigned)ids[row];
    f32x4 x = tab[id * EMBV + lane];
    __builtin_nontemporal_store(x, &out[(unsigned)row * EMBV + lane]);
}

// ---------------------------------------------------------------------------
// CDNA5 async-tensor probe (compiled into the bundle, NOT launched).
// Demonstrates GLOBAL_LOAD_ASYNC_TO_LDS_B128 / GLOBAL_STORE_ASYNC_FROM_LDS_B128
// + s_wait_asynccnt (CDNA5 ISA). Kept separate because an
// LDS-staged copy adds DS bandwidth without raising the HBM roofline.
// ---------------------------------------------------------------------------
__global__ __launch_bounds__(256) void cdna5_async_probe(
    const float* __restrict__ src, float* __restrict__ dst)
{
    __shared__ float buf[256 * 4];
    ((volatile float*)buf)[threadIdx.x] = 0.0f;   // force LDS allocation
    __syncthreads();
#if defined(__HIP_DEVICE_COMPILE__) && defined(__gfx1250__)
    unsigned ldsOff = threadIdx.x * 16u;          // per-lane LDS byte address
    const float* g  = src + threadIdx.x * 4;
    float*       go = dst + threadIdx.x * 4;
    asm volatile("global_load_async_to_lds_b128 %0, %1, off"
                 :: "v"(ldsOff), "v"(g) : "memory");
    asm volatile("s_wait_asynccnt 0" ::: "memory");
    asm volatile("global_store_async_from_lds_b128 %0, %1, off"
                 :: "v"(go), "v"(ldsOff) : "memory");
    asm volatile("s_wait_asynccnt 0" ::: "memory");
#endif
    if (threadIdx.x == 0xFFFFFFFFu) dst[0] = buf[0];  // keep buf live, never taken
}

extern "C" void kernel_launch(void* const* d_in, const int* in_sizes, int n_in,
                              void* d_out, int out_size, void* d_ws, size_t ws_size,
                              hipStream_t stream) {
    const int*   ids = (const int*)d_in[0];          // [8, 65536] int32, flat
    const float* tab = (const float*)d_in[1];        // [1000000, 128] fp32
    float*       out = (float*)d_out;                // [8, 65536, 128] fp32

    const int nRows = in_sizes[0];                   // 524288

    if (nRows % ROWS_PER_BLOCK == 0) {
        const int blocks = nRows / ROWS_PER_BLOCK;   // 8192
        emb_gather_fast<<<blocks, 256, 0, stream>>>(
            ids, (const f32x4*)tab, (f32x4*)out);
    } else {
        const int blocks = (nRows + WAVES_PER_BLOCK - 1) / WAVES_PER_BLOCK;
        emb_gather_generic<<<blocks, 256, 0, stream>>>(
            ids, (const f32x4*)tab, (f32x4*)out, nRows);
    }
}